// BezierAlign_81406810128798
// MI455X (gfx1250) — compile-verified
//
#include <hip/hip_runtime.h>
#include <hip/hip_bf16.h>
#include <stdint.h>

#define OUT_H   16
#define OUT_W   64
#define C_      256
#define H_      160
#define W_      160
#define HW_     (H_ * W_)
#define SCALE_  0.25f

// ---- CDNA5 feature detection (device pass only) ----
#if defined(__HIP_DEVICE_COMPILE__) && __has_builtin(__builtin_amdgcn_tensor_load_to_lds) && __has_builtin(__builtin_amdgcn_s_wait_tensorcnt)
#define HAVE_TDM 1
#else
#define HAVE_TDM 0
#endif

typedef unsigned int v4u __attribute__((ext_vector_type(4)));
typedef int          v8i __attribute__((ext_vector_type(8)));
typedef int          v4i __attribute__((ext_vector_type(4)));

__device__ __forceinline__ float bez(float p0, float p1, float p2, float p3, float t) {
    float mt = 1.0f - t;
    return mt * mt * mt * p0 + 3.0f * t * mt * mt * p1 +
           3.0f * t * t * mt * p2 + t * t * t * p3;
}

// 16 (index, weight) pairs for one (roi, oh, ow): 2x2 samples x 4 bilinear corners.
// Indices are y*W+x into the HxW plane, always clamped in-bounds; invalid samples get w=0.
__device__ void sample_meta(const float* __restrict__ roi, int oh, int ow,
                            int* idx, float* w, int* bidx) {
    float px[8], py[8];
#pragma unroll
    for (int k = 0; k < 8; ++k) {
        px[k] = roi[1 + 2 * k] * SCALE_;
        py[k] = roi[2 + 2 * k] * SCALE_;
    }
    *bidx = (int)roi[0];
    float u = (float)ow / (float)OUT_W;
    float v = (float)oh / (float)OUT_H;
    float x0 = bez(px[0], px[1], px[2], px[3], u);
    float y0 = bez(py[0], py[1], py[2], py[3], u);
    float x1 = bez(px[4], px[5], px[6], px[7], u);
    float y1 = bez(py[4], py[5], py[6], py[7], u);
    float xc = x1 * v + x0 * (1.0f - v) - 0.5f;
    float yc = y1 * v + y0 * (1.0f - v) - 0.5f;
    float roi_w = fmaxf(fabsf(px[0] - px[3]), fabsf(px[4] - px[7]));
    float roi_h = fmaxf(fabsf(py[0] - py[3]), fabsf(py[4] - py[7]));
    float bin_h = roi_h / (float)OUT_H;
    float bin_w = roi_w / (float)OUT_W;
#pragma unroll
    for (int s = 0; s < 4; ++s) {
        int iy = s >> 1, ix = s & 1;
        float yy = yc - 0.5f * bin_h + ((float)iy + 0.5f) * bin_h * 0.5f;
        float xx = xc - 0.5f * bin_w + ((float)ix + 0.5f) * bin_w * 0.5f;
        bool valid = (yy > -1.0f) && (yy < (float)H_) && (xx > -1.0f) && (xx < (float)W_);
        float y = fmaxf(yy, 0.0f), x = fmaxf(xx, 0.0f);
        int yl = min((int)floorf(y), H_ - 1);
        int xl = min((int)floorf(x), W_ - 1);
        int yh = min(yl + 1, H_ - 1);
        int xh = min(xl + 1, W_ - 1);
        if (yl >= H_ - 1) y = (float)yl;
        if (xl >= W_ - 1) x = (float)xl;
        float ly = y - (float)yl, lx = x - (float)xl;
        float hy = 1.0f - ly, hx = 1.0f - lx;
        float vs = valid ? 1.0f : 0.0f;
        idx[4 * s + 0] = yl * W_ + xl;  w[4 * s + 0] = hy * hx * vs;
        idx[4 * s + 1] = yl * W_ + xh;  w[4 * s + 1] = hy * lx * vs;
        idx[4 * s + 2] = yh * W_ + xl;  w[4 * s + 2] = ly * hx * vs;
        idx[4 * s + 3] = yh * W_ + xh;  w[4 * s + 3] = ly * lx * vs;
    }
}

// ---------- Stage 1: NCHW -> NHWC transpose (coalesced both sides via LDS tile) ----------
__global__ __launch_bounds__(256) void nchw_to_nhwc(const float* __restrict__ in,
                                                    float* __restrict__ out) {
    __shared__ float tile[32][33];
    int b = blockIdx.z;
    int cbase = blockIdx.y * 32;
    int pbase = blockIdx.x * 32;
    int tx = threadIdx.x & 31;
    int ty = threadIdx.x >> 5;  // 0..7
    const float* src = in + (size_t)b * C_ * HW_;
#pragma unroll
    for (int k = 0; k < 4; ++k) {
        int c = cbase + ty + k * 8;
        tile[ty + k * 8][tx] = src[(size_t)c * HW_ + (size_t)pbase + tx];
    }
    __syncthreads();
    float* dst = out + (size_t)b * HW_ * C_;
#pragma unroll
    for (int k = 0; k < 4; ++k) {
        int p = pbase + ty + k * 8;
        dst[(size_t)p * C_ + cbase + tx] = tile[tx][ty + k * 8];
    }
}

// ---------- Stage 2: gather + weighted sum from NHWC ----------
// One block per (roi, oh). Lanes across channels; TDM gather-mode DMA stages the
// 16 tap-rows (16 x 1KB) per ow into double-buffered LDS, pipelined on TENSORcnt.
// Output transposed through LDS so stores are ow-contiguous.
__global__ __launch_bounds__(256) void bezier_nhwc(const float* __restrict__ rois,
                                                   const float* __restrict__ nhwc,
                                                   float* __restrict__ out, int R) {
    __shared__ int   sIdx[OUT_W][16];
    __shared__ float sWt[OUT_W][16];
    __shared__ float sT[C_ * 17];          // 16-ow output chunk, padded (17) rows
#if HAVE_TDM
    __shared__ float sBuf[2 * 16 * C_];    // double-buffered 16 rows x 256 ch
#endif
    __shared__ int sB;

    int r   = blockIdx.x >> 4;
    int oh  = blockIdx.x & 15;
    int tid = threadIdx.x;
    if (r >= R) return;

    if (tid < OUT_W) {
        int idx[16]; float w[16]; int b;
        sample_meta(rois + (size_t)r * 17, oh, tid, idx, w, &b);
#pragma unroll
        for (int p = 0; p < 16; ++p) { sIdx[tid][p] = idx[p]; sWt[tid][p] = w[p]; }
        if (tid == 0) sB = b;
    }
    __syncthreads();

    int b = sB;
    const float* base = nhwc + (size_t)b * HW_ * C_;
    size_t obase = (size_t)r * C_ * OUT_H * OUT_W + (size_t)oh * OUT_W;
    int wv = tid >> 5;

#if HAVE_TDM
    // Build a gather-mode D# (16-bit row indices, 16 rows of tile_dim0=256 f32)
    // and issue TENSOR_LOAD_TO_LDS from wave 0 only (EXEC-independent DMA).
    auto issue = [&](int ow) {
        unsigned int lds = (unsigned int)(uintptr_t)&sBuf[(ow & 1) * 16 * C_];
        unsigned long long ga = (unsigned long long)(uintptr_t)base;
        v4u g0; v8i g1; v4i g2, g3;
        g0[0] = 0x80000001u;                               // count=1, gather_mode=1, idx16
        g0[1] = lds;                                       // lds_addr
        g0[2] = (unsigned int)(ga & 0xffffffffull);        // global_addr[31:0]
        g0[3] = (unsigned int)((ga >> 32) & 0x01ffffffull) // global_addr[56:32]
                | (2u << 30);                              // type=2 (image)
        g1[0] = (int)(2u << 16);                           // data_size=4B, mask=0
        g1[1] = (int)((unsigned)C_ << 16);                 // tensor_dim0 = 256
        g1[2] = (int)((unsigned)(HW_ & 0xffff) << 16);     // tensor_dim1 = 25600
        g1[3] = (int)((unsigned)C_ << 16);                 // tile_dim0 = 256
        g1[4] = 16;                                        // tile_dim1 = #indices
        g1[5] = C_;                                        // tensor_dim0_stride = 256
        g1[6] = 0; g1[7] = 0;
        const int* ip = sIdx[ow];
        g2[0] = (ip[0]  & 0xffff) | (ip[1]  << 16);
        g2[1] = (ip[2]  & 0xffff) | (ip[3]  << 16);
        g2[2] = (ip[4]  & 0xffff) | (ip[5]  << 16);
        g2[3] = (ip[6]  & 0xffff) | (ip[7]  << 16);
        g3[0] = (ip[8]  & 0xffff) | (ip[9]  << 16);
        g3[1] = (ip[10] & 0xffff) | (ip[11] << 16);
        g3[2] = (ip[12] & 0xffff) | (ip[13] << 16);
        g3[3] = (ip[14] & 0xffff) | (ip[15] << 16);
        v8i gx = { 0, 0, 0, 0, 0, 0, 0, 0 };               // extra operand (clang-23 6-arg form)
        __builtin_amdgcn_tensor_load_to_lds(g0, g1, g2, g3, gx, 0);
    };
    if (wv == 0) issue(0);
#endif

    for (int ow = 0; ow < OUT_W; ++ow) {
#if HAVE_TDM
        if (wv == 0) {
            if (ow + 1 < OUT_W) { issue(ow + 1); __builtin_amdgcn_s_wait_tensorcnt(1); }
            else                {                __builtin_amdgcn_s_wait_tensorcnt(0); }
        }
        __syncthreads();  // staged taps for ow visible to all waves
#endif
        float acc = 0.0f;
#if HAVE_TDM
        const float* G = &sBuf[(ow & 1) * 16 * C_];
#pragma unroll
        for (int p = 0; p < 16; ++p) acc += sWt[ow][p] * G[p * C_ + tid];
#else
#pragma unroll 4
        for (int p = 0; p < 16; ++p)
            acc += sWt[ow][p] * base[(size_t)sIdx[ow][p] * C_ + tid];
        if (ow + 1 < OUT_W) {  // gfx1250 global_prefetch_b8 for next location's taps
            __builtin_prefetch(&base[(size_t)sIdx[ow + 1][0] * C_ + tid], 0, 1);
            __builtin_prefetch(&base[(size_t)sIdx[ow + 1][8] * C_ + tid], 0, 1);
        }
#endif
        sT[tid * 17 + (ow & 15)] = acc * 0.25f;
        if ((ow & 15) == 15) {
            __syncthreads();
            int owb = ow - 15;
#pragma unroll
            for (int k = 0; k < 16; ++k) {
                int e = k * 256 + tid;
                int c = e >> 4, j = e & 15;
                out[obase + (size_t)c * (OUT_H * OUT_W) + owb + j] = sT[c * 17 + j];
            }
        }
        __syncthreads();  // protect sT reuse and (TDM) staging buffer overwrite
    }
}

// ---------- Fallback: direct NCHW gather (used only if ws is too small) ----------
__global__ __launch_bounds__(256) void bezier_direct(const float* __restrict__ in,
                                                     const float* __restrict__ rois,
                                                     float* __restrict__ out, int R) {
    __shared__ int   sIdx[OUT_W][16];
    __shared__ float sWt[OUT_W][16];
    __shared__ int sB;
    int r   = blockIdx.x >> 4;
    int oh  = blockIdx.x & 15;
    int tid = threadIdx.x;
    if (r >= R) return;
    if (tid < OUT_W) {
        int idx[16]; float w[16]; int b;
        sample_meta(rois + (size_t)r * 17, oh, tid, idx, w, &b);
#pragma unroll
        for (int p = 0; p < 16; ++p) { sIdx[tid][p] = idx[p]; sWt[tid][p] = w[p]; }
        if (tid == 0) sB = b;
    }
    __syncthreads();
    int b = sB;
    int ow = tid & 63, cg = tid >> 6;
    int idx[16]; float w[16];
#pragma unroll
    for (int p = 0; p < 16; ++p) { idx[p] = sIdx[ow][p]; w[p] = sWt[ow][p]; }
    const float* src = in + (size_t)b * C_ * HW_;
    size_t obase = (size_t)r * C_ * OUT_H * OUT_W + (size_t)oh * OUT_W + ow;
    for (int i = 0; i < C_ / 4; ++i) {
        int c = cg * (C_ / 4) + i;
        const float* f = src + (size_t)c * HW_;
        float acc = 0.0f;
#pragma unroll 4
        for (int p = 0; p < 16; ++p) acc += w[p] * f[idx[p]];
        out[obase + (size_t)c * (OUT_H * OUT_W)] = acc * 0.25f;
    }
}

extern "C" void kernel_launch(void* const* d_in, const int* in_sizes, int n_in,
                              void* d_out, int out_size, void* d_ws, size_t ws_size,
                              hipStream_t stream) {
    (void)n_in; (void)out_size;
    const float* input = (const float*)d_in[0];
    const float* rois  = (const float*)d_in[1];
    float* out = (float*)d_out;
    int R = in_sizes[1] / 17;
    int N = in_sizes[0] / (C_ * HW_);
    size_t need = (size_t)N * HW_ * C_ * sizeof(float);
    if (ws_size >= need) {
        dim3 g(HW_ / 32, C_ / 32, N);
        nchw_to_nhwc<<<g, 256, 0, stream>>>(input, (float*)d_ws);
        bezier_nhwc<<<R * OUT_H, 256, 0, stream>>>(rois, (const float*)d_ws, out, R);
    } else {
        bezier_direct<<<R * OUT_H, 256, 0, stream>>>(input, rois, out, R);
    }
}